// Group_46119358824790
// MI455X (gfx1250) — compile-verified
//
#include <hip/hip_runtime.h>
#include <math.h>

// Problem constants (match reference)
#define BATCH 32
#define NPTS  8192
#define NGRP  512   // G
#define KNN   32    // M

typedef float v2f __attribute__((ext_vector_type(2)));
typedef float v8f __attribute__((ext_vector_type(8)));

// ---------------------------------------------------------------------------
// Kernel 1: farthest point sampling. One workgroup (1024 threads) per batch.
// Each thread owns 8 points + 8 running min-distances entirely in registers.
// Per iteration: VALU distance update, wave32 shfl argmax, 32-wave LDS argmax.
// ---------------------------------------------------------------------------
#define FPS_T 1024
#define PPT   (NPTS / FPS_T)   // 8 points per thread

__global__ __launch_bounds__(FPS_T)
void fps_kernel(const float* __restrict__ xyz, float* __restrict__ centers)
{
    const int b = blockIdx.x;
    const float* __restrict__ p = xyz + (size_t)b * NPTS * 3;
    const int t = threadIdx.x;
    const int wave = t >> 5;
    const int lane = t & 31;

    float px[PPT], py[PPT], pz[PPT], md[PPT];
#pragma unroll
    for (int i = 0; i < PPT; ++i) {
        const int n = t + i * FPS_T;              // strided -> coalesced loads
        px[i] = p[3 * n + 0];
        py[i] = p[3 * n + 1];
        pz[i] = p[3 * n + 2];
        md[i] = 1e10f;
    }

    __shared__ float s_bx, s_by, s_bz;
    __shared__ float s_val[32];
    __shared__ int   s_idx[32];

    if (t == 0) {
        // first center is point 0 (deterministic start)
        s_bx = p[0]; s_by = p[1]; s_bz = p[2];
        centers[((size_t)b * NGRP + 0) * 3 + 0] = p[0];
        centers[((size_t)b * NGRP + 0) * 3 + 1] = p[1];
        centers[((size_t)b * NGRP + 0) * 3 + 2] = p[2];
    }
    __syncthreads();

    for (int g = 1; g < NGRP; ++g) {
        const float bx = s_bx, by = s_by, bz = s_bz;

        float bestv = -1.0f;
        int   besti = 0;
#pragma unroll
        for (int i = 0; i < PPT; ++i) {
            const int n = t + i * FPS_T;
            const float dx = px[i] - bx;
            const float dy = py[i] - by;
            const float dz = pz[i] - bz;
            const float d  = fmaf(dx, dx, fmaf(dy, dy, dz * dz));
            md[i] = fminf(md[i], d);
            if (md[i] > bestv) { bestv = md[i]; besti = n; }   // first-max wins
        }

        // wave32 argmax (tie -> lowest index, matching jnp.argmax)
#pragma unroll
        for (int off = 16; off > 0; off >>= 1) {
            const float ov = __shfl_xor(bestv, off, 32);
            const int   oi = __shfl_xor(besti, off, 32);
            if (ov > bestv || (ov == bestv && oi < besti)) { bestv = ov; besti = oi; }
        }
        if (lane == 0) { s_val[wave] = bestv; s_idx[wave] = besti; }
        __syncthreads();

        if (wave == 0) {
            float v = s_val[lane];
            int   i = s_idx[lane];
#pragma unroll
            for (int off = 16; off > 0; off >>= 1) {
                const float ov = __shfl_xor(v, off, 32);
                const int   oi = __shfl_xor(i, off, 32);
                if (ov > v || (ov == v && oi < i)) { v = ov; i = oi; }
            }
            if (lane == 0) {
                const float cx = p[3 * i + 0];
                const float cy = p[3 * i + 1];
                const float cz = p[3 * i + 2];
                s_bx = cx; s_by = cy; s_bz = cz;
                centers[((size_t)b * NGRP + g) * 3 + 0] = cx;
                centers[((size_t)b * NGRP + g) * 3 + 1] = cy;
                centers[((size_t)b * NGRP + g) * 3 + 2] = cz;
            }
        }
        __syncthreads();
    }
}

// ---------------------------------------------------------------------------
// Kernel 2: KNN + gather + center-subtract.
// Block = 256 threads (8 waves), handles one batch b and a tile of 16 centers.
// Squared distance D[m][n] = |c_m|^2 + |p_n|^2 - 2 c_m.p_n is computed as ONE
// V_WMMA_F32_16X16X4_F32 per 16x16 tile:
//   A row m = (-2cx, -2cy, -2cz, |c|^2), B col n = (x, y, z, 1), C = |p_n|^2.
// Top-32 per center kept by 16 owner threads with a worst-tracked list.
// ---------------------------------------------------------------------------
#define T2    256
#define W2    8              // waves per block
#define TG    16             // centers per tile
#define CHUNK (W2 * 16)      // 128 points per round
#define DPAD  4

__global__ __launch_bounds__(T2)
void knn_kernel(const float* __restrict__ xyz,
                float* __restrict__ nbr,          // [B][G][32][3]
                const float* __restrict__ centers, // [B][G][3]
                int* __restrict__ flat_idx)       // [B*G*32]
{
    const int b  = blockIdx.y;
    const int gt = blockIdx.x;              // center tile within batch
    const float* __restrict__ p = xyz + (size_t)b * NPTS * 3;

    const int t    = threadIdx.x;
    const int wave = t >> 5;
    const int lane = t & 31;
    const int m16  = lane & 15;
    const bool hi  = (lane >= 16);

    __shared__ float s_cx[TG], s_cy[TG], s_cz[TG], s_cn2[TG];
    __shared__ float s_d[TG][CHUNK + DPAD];
    __shared__ float s_bd[TG][KNN];
    __shared__ int   s_bi[TG][KNN];
    __shared__ float s_worst[TG];
    __shared__ int   s_wpos[TG];

    if (t < TG) {
        const int gc = gt * TG + t;
        const float cx = centers[((size_t)b * NGRP + gc) * 3 + 0];
        const float cy = centers[((size_t)b * NGRP + gc) * 3 + 1];
        const float cz = centers[((size_t)b * NGRP + gc) * 3 + 2];
        s_cx[t] = cx; s_cy[t] = cy; s_cz[t] = cz;
        s_cn2[t] = fmaf(cx, cx, fmaf(cy, cy, cz * cz));
        s_worst[t] = 3.4e38f;
        s_wpos[t]  = 0;
        for (int j = 0; j < KNN; ++j) { s_bd[t][j] = 3.4e38f; s_bi[t][j] = 0; }
    }
    __syncthreads();

    // A matrix (16x4 f32): lanes 0-15 hold K=0,1 of row m; lanes 16-31 K=2,3.
    v2f a;
    a.x = hi ? (-2.0f * s_cz[m16]) : (-2.0f * s_cx[m16]);
    a.y = hi ? (         s_cn2[m16]) : (-2.0f * s_cy[m16]);

    for (int round = 0; round < NPTS / CHUNK; ++round) {   // 64 rounds
        const int n = round * CHUNK + wave * 16 + m16;     // this lane's column point
        const float x = p[3 * n + 0];
        const float y = p[3 * n + 1];
        const float z = p[3 * n + 2];

        // B matrix (4x16 f32): lanes 0-15 rows K=0,1 -> (x,y); lanes 16-31 rows K=2,3 -> (z,1).
        v2f bb;
        bb.x = hi ? z : x;
        bb.y = hi ? 1.0f : y;

        // C = |p_n|^2 broadcast down the column (same value in all 8 C VGPRs).
        const float q = fmaf(x, x, fmaf(y, y, z * z));
        v8f c;
#pragma unroll
        for (int k = 0; k < 8; ++k) c[k] = q;

        v8f d = __builtin_amdgcn_wmma_f32_16x16x4_f32(
            /*neg_a=*/false, a, /*neg_b=*/false, bb,
            /*c_mod=*/(short)0, c, /*reuse_a=*/false, /*reuse_b=*/false);

        // Scatter D tile to LDS: VGPR v -> row v (lanes 0-15) / row v+8 (lanes 16-31).
        const int col = wave * 16 + m16;
        const int rb  = hi ? 8 : 0;
#pragma unroll
        for (int v = 0; v < 8; ++v) s_d[rb + v][col] = d[v];
        __syncthreads();

        // 16 owner threads merge this 128-point chunk into their top-32 lists.
        if (t < TG) {
            const int m = t;
            float worst = s_worst[m];
            int   wp    = s_wpos[m];
            for (int cidx = 0; cidx < CHUNK; ++cidx) {
                const float dv = s_d[m][cidx];
                if (dv < worst) {
                    s_bd[m][wp] = dv;
                    s_bi[m][wp] = round * CHUNK + cidx;
                    worst = -1.0f;
                    for (int j = 0; j < KNN; ++j)
                        if (s_bd[m][j] > worst) { worst = s_bd[m][j]; wp = j; }
                }
            }
            s_worst[m] = worst;
            s_wpos[m]  = wp;
        }
        __syncthreads();
    }

    // Sort each center's 32 entries ascending by distance (stable by index),
    // matching lax.top_k(-d, k) ordering.
    if (t < TG) {
        const int m = t;
        for (int i = 0; i < KNN - 1; ++i) {
            int best = i;
            for (int j = i + 1; j < KNN; ++j) {
                if (s_bd[m][j] < s_bd[m][best] ||
                    (s_bd[m][j] == s_bd[m][best] && s_bi[m][j] < s_bi[m][best]))
                    best = j;
            }
            if (best != i) {
                const float td = s_bd[m][i]; s_bd[m][i] = s_bd[m][best]; s_bd[m][best] = td;
                const int   ti = s_bi[m][i]; s_bi[m][i] = s_bi[m][best]; s_bi[m][best] = ti;
            }
        }
    }
    __syncthreads();

    // Emit flat_idx and centered neighborhood (512 entries, 2 per thread).
    for (int e = t; e < TG * KNN; e += T2) {
        const int m = e >> 5;
        const int j = e & (KNN - 1);
        const int gc = gt * TG + m;
        const int n  = s_bi[m][j];
        const size_t o = (((size_t)b * NGRP + gc) * KNN + j);
        flat_idx[o] = b * NPTS + n;
        nbr[o * 3 + 0] = p[3 * n + 0] - s_cx[m];
        nbr[o * 3 + 1] = p[3 * n + 1] - s_cy[m];
        nbr[o * 3 + 2] = p[3 * n + 2] - s_cz[m];
    }
}

// ---------------------------------------------------------------------------
// Host launcher. d_out layout (flat, in reference return order):
//   [0 .. B*G*M*3)                      neighborhood  (float32)
//   [B*G*M*3 .. +B*G*3)                 center        (float32)
//   [B*G*M*3+B*G*3 .. +B*G*M)           flat_idx      (int32 bit pattern)
// ---------------------------------------------------------------------------
extern "C" void kernel_launch(void* const* d_in, const int* in_sizes, int n_in,
                              void* d_out, int out_size, void* d_ws, size_t ws_size,
                              hipStream_t stream)
{
    (void)in_sizes; (void)n_in; (void)out_size; (void)d_ws; (void)ws_size;

    const float* xyz = (const float*)d_in[0];
    float* out = (float*)d_out;

    const size_t nbr_elems    = (size_t)BATCH * NGRP * KNN * 3;   // 1572864
    const size_t center_elems = (size_t)BATCH * NGRP * 3;         //   49152

    float* centers  = out + nbr_elems;
    int*   flat_idx = (int*)(out + nbr_elems + center_elems);

    fps_kernel<<<BATCH, FPS_T, 0, stream>>>(xyz, centers);
    knn_kernel<<<dim3(NGRP / TG, BATCH), T2, 0, stream>>>(xyz, out, centers, flat_idx);
}